// GraphConvolution_31593779429782
// MI455X (gfx1250) — compile-verified
//
#include <hip/hip_runtime.h>
#include <math.h>

// ---------------------------------------------------------------------------
// GraphConvolution (e3nn-style, scalar irreps) for MI455X / gfx1250, wave32.
//   out = C_S * fctp(x, a, Wmask) + C_X * fctp(agg, a, Wout)
//   agg = rsqrt(deg) * segsum_dst( nf[src] * eattr * radialMLP(elen) )
//   nf  = rsqrt(deg) * fctp(x, a, Win)
// All GEMMs run on V_WMMA_F32_16X16X4_F32 (exact fp32 matrix pipe).
// ---------------------------------------------------------------------------

typedef float v2f __attribute__((ext_vector_type(2)));
typedef float v8f __attribute__((ext_vector_type(8)));

#define DIM    128   // D
#define NBASIS 10    // NB
#define HID    100   // H

__device__ __forceinline__ v8f wmma_f32_16x16x4(v2f a, v2f b, v8f c) {
  // (neg_a, A, neg_b, B, c_mod, C, reuse_a, reuse_b)
  return __builtin_amdgcn_wmma_f32_16x16x4_f32(false, a, false, b, (short)0, c,
                                               false, false);
}

__constant__ const float INV_SQRT_FAN = 0.08838834764831845f; // 1/sqrt(128*1)
__constant__ const float C_S_K        = 0.3826834323650898f;  // sin(pi/8)
__constant__ const float C_X_K        = 0.9238795325112867f;  // cos(pi/8)
__constant__ const float RSQRT_NB     = 0.31622776601683794f; // 1/sqrt(10)
__constant__ const float RSQRT_H      = 0.1f;                 // 1/sqrt(100)

// ---------------------------------------------------------------------------
// Kernel A: nf = (x @ Win) * attr * rsqrt(deg) * INV_SQRT_FAN
//           out = C_S * (x @ Wmask) * attr * INV_SQRT_FAN
// Workgroup: 256 threads = 8 waves; 16 node rows x 128 output cols.
// ---------------------------------------------------------------------------
__global__ void __launch_bounds__(256)
node_linear_in_kernel(const float* __restrict__ x,
                      const float* __restrict__ attr,
                      const float* __restrict__ deg,
                      const float* __restrict__ Win,
                      const float* __restrict__ Wmask,
                      float* __restrict__ nf,
                      float* __restrict__ outMask,
                      int N)
{
  __shared__ float xs[16 * 129];   // stride 129 -> conflict-free column reads
  __shared__ float sF[16];
  __shared__ float sM[16];

  const int tid = threadIdx.x;
  const int rowBase = blockIdx.x * 16;

  for (int idx = tid; idx < 16 * 128; idx += 256) {
    int r = idx >> 7, c = idx & 127;
    int gr = rowBase + r;
    xs[r * 129 + c] = (gr < N) ? x[(size_t)gr * DIM + c] : 0.0f;
  }
  if (tid < 16) {
    int gr = rowBase + tid;
    float a = (gr < N) ? attr[gr] : 0.0f;
    float d = (gr < N) ? deg[gr] : 1.0f;
    sF[tid] = INV_SQRT_FAN * a * rsqrtf(d);
    sM[tid] = C_S_K * INV_SQRT_FAN * a;
  }
  __syncthreads();

  const int lane    = tid & 31;
  const int wave    = tid >> 5;
  const int colBase = wave * 16;       // 8 waves x 16 cols = 128
  const int mrow    = lane & 15;       // A: M index / B,C: N index
  const int khalf   = (lane >> 4) * 2; // lanes 16..31 carry K+2,K+3

  v8f accF = {};
  v8f accM = {};
  const int n = colBase + mrow;
  for (int k = 0; k < DIM; k += 4) {
    v2f a;
    a.x = xs[mrow * 129 + k + khalf];
    a.y = xs[mrow * 129 + k + khalf + 1];
    v2f bF, bM;
    bF.x = Win  [(size_t)(k + khalf)     * DIM + n];
    bF.y = Win  [(size_t)(k + khalf + 1) * DIM + n];
    bM.x = Wmask[(size_t)(k + khalf)     * DIM + n];
    bM.y = Wmask[(size_t)(k + khalf + 1) * DIM + n];
    accF = wmma_f32_16x16x4(a, bF, accF);
    accM = wmma_f32_16x16x4(a, bM, accM);
  }

  const int roff = (lane >> 4) * 8;
  const int col  = colBase + (lane & 15);
  for (int i = 0; i < 8; ++i) {
    int r  = i + roff;
    int gr = rowBase + r;
    if (gr < N) {
      nf     [(size_t)gr * DIM + col] = accF[i] * sF[r];
      outMask[(size_t)gr * DIM + col] = accM[i] * sM[r];
    }
  }
}

// ---------------------------------------------------------------------------
// Kernel B: per-edge radial MLP (WMMA) + gather * attr * weight + scatter-add.
// Workgroup = 128 threads = 4 waves; each wave owns 32 edges (two M-tiles
// sharing every B fragment -> 2 WMMAs per B load).
//   h = silu([32x10] @ W0) * const scales folded into A-frags / eattr.
// GEMM2 processes one 16-col output tile at a time and consumes it.
// ---------------------------------------------------------------------------
__global__ void __launch_bounds__(128)
edge_kernel(const int*   __restrict__ esrc,
            const int*   __restrict__ edst,
            const float* __restrict__ eattr,
            const float* __restrict__ elen,   // [E, NBASIS]
            const float* __restrict__ W0,     // [NBASIS, HID]
            const float* __restrict__ W1,     // [HID, DIM]
            const float* __restrict__ nf,     // [N, DIM]
            float*       __restrict__ agg,    // [N, DIM]
            int E)
{
  __shared__ float hbuf[4][32 * 113];  // per-wave h, odd stride = no conflicts
  __shared__ float w0p[12 * 112];      // zero-padded W0 (K 10->12, N 100->112)
  __shared__ int   ssrc[4][32];
  __shared__ int   sdst[4][32];
  __shared__ float sattr[4][32];       // eattr * RSQRT_H folded in

  const int tid   = threadIdx.x;
  const int lane  = tid & 31;
  const int wave  = tid >> 5;
  const int ebase = (blockIdx.x * 4 + wave) * 32;
  float* h = hbuf[wave];

  // stage zero-padded W0 once per workgroup (guards leave the inner loop)
  for (int idx = tid; idx < 12 * 112; idx += 128) {
    int k = idx / 112, n = idx - k * 112;
    w0p[idx] = (k < NBASIS && n < HID) ? W0[(size_t)k * HID + n] : 0.0f;
  }
  // stage per-edge scalars (one edge per lane)
  {
    int e  = ebase + lane;
    int ec = (e < E) ? e : (E > 0 ? E - 1 : 0);
    ssrc[wave][lane]  = esrc[ec];
    sdst[wave][lane]  = edst[ec];
    sattr[wave][lane] = (e < E) ? eattr[ec] * RSQRT_H : 0.0f;
  }
  __syncthreads();

  const int mrow  = lane & 15;
  const int khalf = (lane >> 4) * 2;
  const int ncol  = lane & 15;
  const int roff  = (lane >> 4) * 8;

  // --- A fragments for GEMM1, both M-tiles, RSQRT_NB folded in -------------
  v2f a1[2][3];
  for (int t = 0; t < 2; ++t) {
    int e = ebase + t * 16 + mrow;
    bool ev = (e < E);
    const float* erow = elen + (size_t)(ev ? e : 0) * NBASIS;
    for (int ks = 0; ks < 3; ++ks) {
      int k0 = ks * 4 + khalf;
      a1[t][ks].x = (ev && (k0     < NBASIS)) ? erow[k0]     * RSQRT_NB : 0.0f;
      a1[t][ks].y = (ev && (k0 + 1 < NBASIS)) ? erow[k0 + 1] * RSQRT_NB : 0.0f;
    }
  }

  // --- GEMM1 + silu -> h (N = 100 padded to 112 -> 7 tiles) ----------------
  for (int nt = 0; nt < 7; ++nt) {
    int n = nt * 16 + ncol;
    v8f acc0 = {};
    v8f acc1 = {};
    for (int ks = 0; ks < 3; ++ks) {
      int k0 = ks * 4 + khalf;
      v2f b;
      b.x = w0p[k0 * 112 + n];
      b.y = w0p[(k0 + 1) * 112 + n];
      acc0 = wmma_f32_16x16x4(a1[0][ks], b, acc0);
      acc1 = wmma_f32_16x16x4(a1[1][ks], b, acc1);
    }
    for (int i = 0; i < 8; ++i) {
      float v0 = acc0[i];
      float v1 = acc1[i];
      v0 = v0 / (1.0f + __expf(-v0));   // silu
      v1 = v1 / (1.0f + __expf(-v1));
      h[(i + roff) * 113 + n]      = v0;
      h[(16 + i + roff) * 113 + n] = v1;
    }
  }

  // --- GEMM2: [32x100] @ [100x128], one 16-col tile at a time --------------
  for (int nt = 0; nt < 8; ++nt) {
    int n = nt * 16 + ncol;
    v8f acc0 = {};
    v8f acc1 = {};
    for (int ks = 0; ks < 25; ++ks) {
      int k0 = ks * 4 + khalf;
      v2f a0, a1v;
      a0.x  = h[mrow * 113 + k0];
      a0.y  = h[mrow * 113 + k0 + 1];
      a1v.x = h[(16 + mrow) * 113 + k0];
      a1v.y = h[(16 + mrow) * 113 + k0 + 1];
      v2f b;
      b.x = W1[(size_t) k0      * DIM + n];
      b.y = W1[(size_t)(k0 + 1) * DIM + n];
      acc0 = wmma_f32_16x16x4(a0,  b, acc0);
      acc1 = wmma_f32_16x16x4(a1v, b, acc1);
    }
    // consume: msg = nf[src] * (eattr/sqrt(H)) * edge_weight ; scatter-add
    for (int t = 0; t < 2; ++t) {
      const v8f& acc = t ? acc1 : acc0;
      for (int i = 0; i < 8; ++i) {
        int m = t * 16 + i + roff;
        int e = ebase + m;
        if (e < E) {
          int   s  = ssrc[wave][m];
          int   d  = sdst[wave][m];
          float ms = nf[(size_t)s * DIM + n] * sattr[wave][m] * acc[i];
          atomicAdd(&agg[(size_t)d * DIM + n], ms);
        }
      }
    }
  }
}

// ---------------------------------------------------------------------------
// Kernel C: out += C_X * ((agg * rsqrt(deg)) @ Wout) * attr * INV_SQRT_FAN
// ---------------------------------------------------------------------------
__global__ void __launch_bounds__(256)
node_linear_out_kernel(const float* __restrict__ agg,
                       const float* __restrict__ attr,
                       const float* __restrict__ deg,
                       const float* __restrict__ Wout,
                       float* __restrict__ out,
                       int N)
{
  __shared__ float xs[16 * 129];
  __shared__ float sO[16];

  const int tid = threadIdx.x;
  const int rowBase = blockIdx.x * 16;

  // pre-scale rows by rsqrt(deg) while staging
  for (int idx = tid; idx < 16 * 128; idx += 256) {
    int r = idx >> 7, c = idx & 127;
    int gr = rowBase + r;
    float rs = (gr < N) ? rsqrtf(deg[gr]) : 0.0f;
    xs[r * 129 + c] = (gr < N) ? agg[(size_t)gr * DIM + c] * rs : 0.0f;
  }
  if (tid < 16) {
    int gr = rowBase + tid;
    float a = (gr < N) ? attr[gr] : 0.0f;
    sO[tid] = C_X_K * INV_SQRT_FAN * a;
  }
  __syncthreads();

  const int lane    = tid & 31;
  const int wave    = tid >> 5;
  const int colBase = wave * 16;
  const int mrow    = lane & 15;
  const int khalf   = (lane >> 4) * 2;

  v8f acc = {};
  const int n = colBase + mrow;
  for (int k = 0; k < DIM; k += 4) {
    v2f a;
    a.x = xs[mrow * 129 + k + khalf];
    a.y = xs[mrow * 129 + k + khalf + 1];
    v2f b;
    b.x = Wout[(size_t)(k + khalf)     * DIM + n];
    b.y = Wout[(size_t)(k + khalf + 1) * DIM + n];
    acc = wmma_f32_16x16x4(a, b, acc);
  }

  const int roff = (lane >> 4) * 8;
  const int col  = colBase + (lane & 15);
  for (int i = 0; i < 8; ++i) {
    int r  = i + roff;
    int gr = rowBase + r;
    if (gr < N) {
      size_t idx = (size_t)gr * DIM + col;
      out[idx] = out[idx] + acc[i] * sO[r];   // out already holds C_S * mask
    }
  }
}

// ---------------------------------------------------------------------------
extern "C" void kernel_launch(void* const* d_in, const int* in_sizes, int n_in,
                              void* d_out, int out_size, void* d_ws, size_t ws_size,
                              hipStream_t stream)
{
  const float* node_input = (const float*)d_in[0];   // [N,128]
  const float* node_attr  = (const float*)d_in[1];   // [N,1]
  const float* node_deg   = (const float*)d_in[2];   // [N,1]
  const int*   edge_src   = (const int*)  d_in[3];   // [E]
  const int*   edge_dst   = (const int*)  d_in[4];   // [E]
  const float* edge_attr  = (const float*)d_in[5];   // [E,1]
  const float* elen       = (const float*)d_in[6];   // [E,10]
  const float* w_lin_in   = (const float*)d_in[7];   // [128,1,128]
  const float* w_lin_mask = (const float*)d_in[8];   // [128,1,128]
  const float* w_lin_out  = (const float*)d_in[9];   // [128,1,128]
  const float* W0         = (const float*)d_in[10];  // [10,100]
  const float* W1         = (const float*)d_in[11];  // [100,128]

  const int N = in_sizes[2];   // node_deg has N elements
  const int E = in_sizes[3];

  float* out = (float*)d_out;
  float* nf  = (float*)d_ws;                       // [N,128] node_features
  float* agg = nf + (size_t)N * DIM;               // [N,128] segment sum

  hipMemsetAsync(agg, 0, (size_t)N * DIM * sizeof(float), stream);

  const int nodeBlocks = (N + 15) / 16;
  node_linear_in_kernel<<<nodeBlocks, 256, 0, stream>>>(
      node_input, node_attr, node_deg, w_lin_in, w_lin_mask, nf, out, N);

  const int edgeBlocks = (E + 127) / 128;           // 4 waves x 32 edges
  edge_kernel<<<edgeBlocks, 128, 0, stream>>>(
      edge_src, edge_dst, edge_attr, elen, W0, W1, nf, agg, E);

  node_linear_out_kernel<<<nodeBlocks, 256, 0, stream>>>(
      agg, node_attr, node_deg, w_lin_out, out, N);
}